// ConnectedLossV2_83760452206647
// MI455X (gfx1250) — compile-verified
//
#include <hip/hip_runtime.h>
#include <hip/hip_bf16.h>
#include <math.h>

typedef float v2f __attribute__((ext_vector_type(2)));
typedef float v8f __attribute__((ext_vector_type(8)));

#define H_DIM 512
#define W_DIM 512
#define HW_C  (H_DIM * W_DIM)
#define TMAXL 16
#define CCL_ITERS 128

#define TILE  32
#define HALO  8
#define LSIZE (TILE + 2 * HALO)     /* 48 */
#define LAREA (LSIZE * LSIZE)       /* 2304 */
#define STEPS_PER_LAUNCH 8
#define CCL_LAUNCHES (CCL_ITERS / STEPS_PER_LAUNCH)  /* 16, even */

#define SP0_C 0.6931471805599453f   /* softplus(0) */
#define SP1_C 1.3132616875182228f   /* softplus(1) */
#define S1_C  0.7310585786300049f   /* sigmoid(1)  */

// stats word layout (unsigned words):
// [0..3]  float sums: s0=bce-term, s1=sum(p), s2=sum(y), s3=sum(p*y)
// [4..20] uint cnt_t[17]
// [21]    uint n_pred
// [22]    float acc, [23] float matched, [24] float un_tgt
// [26..27] u64 argmin candidate (8-byte aligned)

__device__ __forceinline__ float softplusf(float x) {
  return fmaxf(x, 0.0f) + log1pf(expf(-fabsf(x)));
}

__global__ void k_init(const float* __restrict__ pred, int N,
                       int* __restrict__ lab0, unsigned* __restrict__ cnt_p,
                       unsigned* __restrict__ inter_t, unsigned char* __restrict__ used,
                       unsigned char* __restrict__ fg, unsigned* __restrict__ stats) {
  int i = blockIdx.x * blockDim.x + threadIdx.x;
  if (i < 32) stats[i] = 0u;
  if (i >= N) return;
  int b  = i >> 18;            // i / HW  (HW = 2^18)
  int hw = i & (HW_C - 1);
  float p0 = pred[(size_t)(b * 2)     * HW_C + hw];
  float p1 = pred[(size_t)(b * 2 + 1) * HW_C + hw];
  bool f = p1 > p0;            // argmax over 2 channels (first wins ties)
  fg[i]   = f ? 1 : 0;
  lab0[i] = f ? (i + 1) : 0;
  cnt_p[i]   = 0u;
  inter_t[i] = 0u;
  used[i]    = 0;
}

// Halo-tiled max-label propagation: 8 global-equivalent Jacobi steps per launch,
// computed entirely in LDS. Interior 32x32 written back is bit-exact vs 8
// synchronous global sweeps (stale halo advances 1 cell/step, interior is >=8
// cells from the LDS edge).
__global__ void k_ccl_tiled(const int* __restrict__ src, int* __restrict__ dst,
                            const unsigned char* __restrict__ fg) {
  __shared__ int lA[LAREA];
  __shared__ int lB[LAREA];
  __shared__ unsigned char lm[LAREA];

  const int tid = threadIdx.x;
  const int x0 = blockIdx.x * TILE - HALO;
  const int y0 = blockIdx.y * TILE - HALO;
  const int b  = blockIdx.z;
  const int base = b * HW_C;

  // load 48x48 region (2304 = 9 * 256 cells), OOB -> 0 / mask 0
  for (int k = tid; k < LAREA; k += 256) {
    int ly = k / LSIZE, lx = k - ly * LSIZE;
    int gy = y0 + ly, gx = x0 + lx;
    int v = 0; unsigned char mm = 0;
    if (gy >= 0 && gy < H_DIM && gx >= 0 && gx < W_DIM) {
      int i = base + gy * W_DIM + gx;
      v = src[i]; mm = fg[i];
    }
    lA[k] = v; lB[k] = v; lm[k] = mm;
  }
  __syncthreads();

  int* cur = lA;
  int* nxt = lB;
  for (int s = 0; s < STEPS_PER_LAUNCH; ++s) {
    for (int k = tid; k < (LSIZE - 2) * (LSIZE - 2); k += 256) {
      int ly = k / (LSIZE - 2) + 1;
      int lx = k - (ly - 1) * (LSIZE - 2) + 1;
      int idx = ly * LSIZE + lx;
      int v = 0;
      if (lm[idx]) {
        v = cur[idx];
        v = max(v, cur[idx - LSIZE]);
        v = max(v, cur[idx + LSIZE]);
        v = max(v, cur[idx - 1]);
        v = max(v, cur[idx + 1]);
      }
      nxt[idx] = v;
    }
    __syncthreads();
    int* t = cur; cur = nxt; nxt = t;
  }

  // write back 32x32 interior (1024 = 4 * 256 cells)
  for (int k = tid; k < TILE * TILE; k += 256) {
    int ly = HALO + (k >> 5), lx = HALO + (k & 31);
    int gy = y0 + ly, gx = x0 + lx;          // always in-image
    dst[base + gy * W_DIM + gx] = cur[ly * LSIZE + lx];
  }
}

// WMMA-based 4-statistic global reduction + cnt_p scatter + target histogram.
__global__ void k_stats(const float* __restrict__ pred, const int* __restrict__ tgt,
                        const unsigned char* __restrict__ fg, const int* __restrict__ lab,
                        unsigned* __restrict__ cnt_p, unsigned* __restrict__ stats, int N) {
  __shared__ unsigned lhist[TMAXL + 1];
  if (threadIdx.x <= TMAXL) lhist[threadIdx.x] = 0u;
  __syncthreads();

  int tid = blockIdx.x * blockDim.x + threadIdx.x;
  int T   = gridDim.x * blockDim.x;
  int rounds = (N + 2 * T - 1) / (2 * T);   // uniform across the wave

  v2f ones = {1.0f, 1.0f};
  v8f c0 = {}, c1 = {}, c2 = {}, c3 = {};

  for (int r = 0; r < rounds; ++r) {
    float sA0 = 0.f, sA1 = 0.f, sA2 = 0.f, sA3 = 0.f;
    float sB0 = 0.f, sB1 = 0.f, sB2 = 0.f, sB3 = 0.f;
    int ia = tid + (2 * r) * T;
    int ib = tid + (2 * r + 1) * T;
    if (ia < N) {
      int b = ia >> 18, hw = ia & (HW_C - 1);
      float p1 = pred[(size_t)(b * 2 + 1) * HW_C + hw];
      bool f = fg[ia] != 0;
      float z = f ? p1 : 0.0f;
      int t = tgt[ia]; t = t < 0 ? 0 : (t > TMAXL ? TMAXL : t);
      float y  = (t > 0) ? 1.0f : 0.0f;
      float pr = 1.0f / (1.0f + expf(-z));
      sA0 = softplusf(z) - z * y; sA1 = pr; sA2 = y; sA3 = pr * y;
      atomicAdd(&lhist[t], 1u);
      if (f) atomicAdd(&cnt_p[lab[ia] - 1], 1u);
    }
    if (ib < N) {
      int b = ib >> 18, hw = ib & (HW_C - 1);
      float p1 = pred[(size_t)(b * 2 + 1) * HW_C + hw];
      bool f = fg[ib] != 0;
      float z = f ? p1 : 0.0f;
      int t = tgt[ib]; t = t < 0 ? 0 : (t > TMAXL ? TMAXL : t);
      float y  = (t > 0) ? 1.0f : 0.0f;
      float pr = 1.0f / (1.0f + expf(-z));
      sB0 = softplusf(z) - z * y; sB1 = pr; sB2 = y; sB3 = pr * y;
      atomicAdd(&lhist[t], 1u);
      if (f) atomicAdd(&cnt_p[lab[ib] - 1], 1u);
    }
    // converged here: EXEC is all-ones for the WMMA issues.
    v2f b0 = {sA0, sB0}, b1 = {sA1, sB1}, b2 = {sA2, sB2}, b3 = {sA3, sB3};
    // D = ones(16x4) x B(4x16) + C : accumulates column sums of B into every row of C.
    c0 = __builtin_amdgcn_wmma_f32_16x16x4_f32(false, ones, false, b0, (short)0, c0, false, false);
    c1 = __builtin_amdgcn_wmma_f32_16x16x4_f32(false, ones, false, b1, (short)0, c1, false, false);
    c2 = __builtin_amdgcn_wmma_f32_16x16x4_f32(false, ones, false, b2, (short)0, c2, false, false);
    c3 = __builtin_amdgcn_wmma_f32_16x16x4_f32(false, ones, false, b3, (short)0, c3, false, false);
  }

  // C VGPR0: lanes 0-15 hold row M=0, lanes 16-31 hold row M=8; all rows equal
  // (A all-ones), so the 32-lane sum double-counts the 16 column totals -> *0.5.
  float v0 = c0[0], v1 = c1[0], v2 = c2[0], v3 = c3[0];
  for (int o = 16; o > 0; o >>= 1) {
    v0 += __shfl_xor(v0, o);
    v1 += __shfl_xor(v1, o);
    v2 += __shfl_xor(v2, o);
    v3 += __shfl_xor(v3, o);
  }
  if ((threadIdx.x & 31) == 0) {
    float* fs = (float*)stats;
    atomicAdd(&fs[0], v0 * 0.5f);
    atomicAdd(&fs[1], v1 * 0.5f);
    atomicAdd(&fs[2], v2 * 0.5f);
    atomicAdd(&fs[3], v3 * 0.5f);
  }
  __syncthreads();
  if (threadIdx.x <= TMAXL && lhist[threadIdx.x]) atomicAdd(&stats[4 + threadIdx.x], lhist[threadIdx.x]);
}

__global__ void k_npred(const unsigned* __restrict__ cnt_p, unsigned* __restrict__ stats, int N) {
  int p = blockIdx.x * blockDim.x + threadIdx.x;
  bool nz = (p < N) && (cnt_p[p] > 0u);
  unsigned long long m = __ballot(nz);
  if ((threadIdx.x & 31) == 0) {
    unsigned c = (unsigned)__popcll(m);
    if (c) atomicAdd(&stats[21], c);
  }
}

__global__ void k_clear(unsigned* __restrict__ inter_t, unsigned long long* __restrict__ cand, int N) {
  int i = blockIdx.x * blockDim.x + threadIdx.x;
  if (i == 0) *cand = ~0ull;
  if (i < N) inter_t[i] = 0u;
}

__global__ void k_inter(const unsigned char* __restrict__ fg, const int* __restrict__ tgt,
                        const int* __restrict__ lab, unsigned* __restrict__ inter_t, int t, int N) {
  int i = blockIdx.x * blockDim.x + threadIdx.x;
  if (i >= N) return;
  if (fg[i] && tgt[i] == t) atomicAdd(&inter_t[lab[i] - 1], 1u);
}

__global__ void k_argmin(const unsigned* __restrict__ cnt_p, const unsigned* __restrict__ inter_t,
                         const unsigned char* __restrict__ used, const unsigned* __restrict__ stats,
                         unsigned long long* __restrict__ cand, int t, int N, float Nf) {
  int p = blockIdx.x * blockDim.x + threadIdx.x;
  unsigned long long key = ~0ull;
  if (p < N) {
    unsigned cp = cnt_p[p];
    if (cp > 0u && used[p] == 0) {
      float ct  = (float)stats[4 + t];
      float ip  = (float)inter_t[p];
      float cpf = (float)cp;
      float bce    = (Nf * SP0_C + cpf * (SP1_C - SP0_C) - ip) / Nf;
      float sum_pt = 0.5f * ct + (S1_C - 0.5f) * ip;
      float sum_p  = 0.5f * Nf + (S1_C - 0.5f) * cpf;
      float dice   = 1.0f - (2.0f * sum_pt + 1.0f) / (sum_p + ct + 1.0f);
      float loss   = bce + dice;                  // provably >= 0
      key = ((unsigned long long)__float_as_uint(loss) << 32) | (unsigned)p;
    }
  }
  for (int o = 16; o > 0; o >>= 1) {
    unsigned long long other = __shfl_xor(key, o);
    key = (other < key) ? other : key;
  }
  if ((threadIdx.x & 31) == 0 && key != ~0ull) atomicMin(cand, key);
}

__global__ void k_finalize(unsigned* __restrict__ stats, unsigned char* __restrict__ used,
                           const unsigned long long* __restrict__ cand, int t) {
  if (blockIdx.x != 0 || threadIdx.x != 0) return;
  float* fs = (float*)stats;
  unsigned long long cd = *cand;
  bool present = stats[4 + t] > 0u;
  bool any     = cd != ~0ull;
  if (present && any) {
    fs[22] += __uint_as_float((unsigned)(cd >> 32));   // acc += min loss
    used[(unsigned)(cd & 0xFFFFFFFFu)] = 1;
    fs[23] += 1.0f;                                    // matched
  } else if (present) {
    fs[24] += 1.0f;                                    // un_tgt
  }
}

__global__ void k_final(const unsigned* __restrict__ stats, float* __restrict__ out, float Nf) {
  if (blockIdx.x != 0 || threadIdx.x != 0) return;
  const float* fs = (const float*)stats;
  float bce  = fs[0] / Nf;
  float dice = 1.0f - (2.0f * fs[3] + 1.0f) / (fs[1] + fs[2] + 1.0f);
  float res  = bce + dice + fs[22] + ((float)stats[21] - fs[23]) + fs[24];
  out[0] = res;
}

extern "C" void kernel_launch(void* const* d_in, const int* in_sizes, int n_in,
                              void* d_out, int out_size, void* d_ws, size_t ws_size,
                              hipStream_t stream) {
  const float* pred = (const float*)d_in[0];   // [B,2,512,512] f32
  const int*   tgt  = (const int*)d_in[1];     // [B,1,512,512] i32
  float* out = (float*)d_out;

  const int N  = in_sizes[1];                  // B*H*W
  const int B  = N / HW_C;
  const float Nf = (float)N;

  char* ws = (char*)d_ws;
  int*            lab0    = (int*)ws;
  int*            lab1    = (int*)(ws + (size_t)N * 4);
  unsigned*       cnt_p   = (unsigned*)(ws + (size_t)N * 8);
  unsigned*       inter_t = (unsigned*)(ws + (size_t)N * 12);
  unsigned char*  used    = (unsigned char*)(ws + (size_t)N * 16);
  unsigned char*  fg      = (unsigned char*)(ws + (size_t)N * 17);
  unsigned*       stats   = (unsigned*)(ws + (size_t)N * 18);   // 128 B, 8-aligned
  unsigned long long* cand = (unsigned long long*)(stats + 26);

  const int threads = 256;
  const int blocks  = (N + threads - 1) / threads;

  k_init<<<blocks, threads, 0, stream>>>(pred, N, lab0, cnt_p, inter_t, used, fg, stats);

  // 128 propagation iterations = 16 launches x 8 LDS-resident steps.
  dim3 cgrid(W_DIM / TILE, H_DIM / TILE, B);
  for (int it = 0; it < CCL_LAUNCHES; ++it) {
    const int* s = (it & 1) ? lab1 : lab0;
    int*       d = (it & 1) ? lab0 : lab1;
    k_ccl_tiled<<<cgrid, threads, 0, stream>>>(s, d, fg);
  }
  // CCL_LAUNCHES is even -> result resides in lab0.

  k_stats<<<512, threads, 0, stream>>>(pred, tgt, fg, lab0, cnt_p, stats, N);
  k_npred<<<blocks, threads, 0, stream>>>(cnt_p, stats, N);

  for (int t = 1; t <= TMAXL; ++t) {
    k_clear<<<blocks, threads, 0, stream>>>(inter_t, cand, N);
    k_inter<<<blocks, threads, 0, stream>>>(fg, tgt, lab0, inter_t, t, N);
    k_argmin<<<blocks, threads, 0, stream>>>(cnt_p, inter_t, used, stats, cand, t, N, Nf);
    k_finalize<<<1, 1, 0, stream>>>(stats, used, cand, t);
  }

  k_final<<<1, 1, 0, stream>>>(stats, out, Nf);
}